// GNN_21801253995179
// MI455X (gfx1250) — compile-verified
//
#include <hip/hip_runtime.h>

typedef __attribute__((ext_vector_type(2))) float v2f;
typedef __attribute__((ext_vector_type(8))) float v8f;

#define F1 128
#define F2 64

// ---------------- utility: zero a float buffer ----------------
__global__ void gnn_zero(float* __restrict__ p, int n) {
  int i = blockIdx.x * blockDim.x + threadIdx.x;
  int stride = gridDim.x * blockDim.x;
  for (int j = i; j < n; j += stride) p[j] = 0.0f;
}

// ---------------- degrees: xdeg[row]++ , deg[col]++ ----------------
__global__ void gnn_degrees(const int* __restrict__ row, const int* __restrict__ col,
                            float* __restrict__ deg, float* __restrict__ xdeg, int E) {
  int i = blockIdx.x * blockDim.x + threadIdx.x;
  int stride = gridDim.x * blockDim.x;
  for (int e = i; e < E; e += stride) {
    unsafeAtomicAdd(&xdeg[row[e]], 1.0f);
    unsafeAtomicAdd(&deg[col[e]], 1.0f);
  }
}

// ---------------- dinv[n] = rsqrt(indeg[n] + 1)  (self-loop) ----------------
__global__ void gnn_dinv(float* __restrict__ deg, int N) {
  int i = blockIdx.x * blockDim.x + threadIdx.x;
  if (i < N) deg[i] = rsqrtf(deg[i] + 1.0f);   // deg+1 >= 1, never zero
}

// ---------------- layer-1 collapsed to scalar scatter ----------------
// s[c] += dinv[r]*dinv[c]*xdeg[r]
__global__ void gnn_scatter_s(const int* __restrict__ row, const int* __restrict__ col,
                              const float* __restrict__ dinv, const float* __restrict__ xdeg,
                              float* __restrict__ s, int E) {
  int i = blockIdx.x * blockDim.x + threadIdx.x;
  int stride = gridDim.x * blockDim.x;
  for (int e = i; e < E; e += stride) {
    int r = row[e], c = col[e];
    unsafeAtomicAdd(&s[c], dinv[r] * dinv[c] * xdeg[r]);
  }
}

// self-loop term: s[n] += dinv[n]^2 * xdeg[n]
__global__ void gnn_finalize_s(const float* __restrict__ dinv, const float* __restrict__ xdeg,
                               float* __restrict__ s, int N) {
  int i = blockIdx.x * blockDim.x + threadIdx.x;
  if (i < N) s[i] += dinv[i] * dinv[i] * xdeg[i];
}

// ---------------- fused h2 = relu(s (x) W1 + b1) @ W2, f32 WMMA ----------------
// Block = 256 threads = 8 waves; each wave owns a 16-row tile x full 64 cols.
// A fragment (16x4 f32): lane m = lane&15, VGPR0/1 hold K = koff, koff+1
// where koff = (lane>=16) ? 2 : 0 (ISA 7.12.2).  A is generated on the fly
// from the per-row scalar s: a[m,k] = relu(s[m]*W1[k] + b1[k]).
__global__ __launch_bounds__(256) void gnn_gemm_wmma(
    const float* __restrict__ s, const float* __restrict__ W1, const float* __restrict__ b1,
    const float* __restrict__ W2, float* __restrict__ h2, int N) {
  __shared__ float lW2[F1 * F2];  // 32 KB of the 320 KB WGP LDS
  __shared__ float lW1[F1];
  __shared__ float lb1[F1];

  int t = threadIdx.x;
  for (int i = t; i < F1 * F2; i += 256) lW2[i] = W2[i];
  if (t < F1) { lW1[t] = W1[t]; lb1[t] = b1[t]; }
  __syncthreads();

  int wave = t >> 5;
  int lane = t & 31;
  int m    = lane & 15;
  int koff = (lane >> 4) << 1;           // 0 or 2
  int tile = blockIdx.x * 8 + wave;      // 16-row tile index
  int rowA = tile * 16 + m;
  float sv = (rowA < N) ? s[rowA] : 0.0f;

  v8f c0 = {}, c1 = {}, c2 = {}, c3 = {};
  int n0 = lane & 15;                    // B/D column within a 16-col tile

#pragma unroll
  for (int k0 = 0; k0 < F1; k0 += 4) {
    int ka = k0 + koff;
    float a0 = sv * lW1[ka]     + lb1[ka];
    float a1 = sv * lW1[ka + 1] + lb1[ka + 1];
    v2f a;
    a.x = a0 > 0.0f ? a0 : 0.0f;        // relu: layer-1 activation fused in
    a.y = a1 > 0.0f ? a1 : 0.0f;

    v2f b;
    b.x = lW2[ka * F2 + n0];
    b.y = lW2[(ka + 1) * F2 + n0];
    c0 = __builtin_amdgcn_wmma_f32_16x16x4_f32(false, a, false, b, (short)0, c0, false, false);
    b.x = lW2[ka * F2 + 16 + n0];
    b.y = lW2[(ka + 1) * F2 + 16 + n0];
    c1 = __builtin_amdgcn_wmma_f32_16x16x4_f32(false, a, false, b, (short)0, c1, false, false);
    b.x = lW2[ka * F2 + 32 + n0];
    b.y = lW2[(ka + 1) * F2 + 32 + n0];
    c2 = __builtin_amdgcn_wmma_f32_16x16x4_f32(false, a, false, b, (short)0, c2, false, false);
    b.x = lW2[ka * F2 + 48 + n0];
    b.y = lW2[(ka + 1) * F2 + 48 + n0];
    c3 = __builtin_amdgcn_wmma_f32_16x16x4_f32(false, a, false, b, (short)0, c3, false, false);
  }

  // D layout: VGPR v holds M = v + 8*(lane>=16), N = lane&15 (ISA 7.12.2)
  int mbase = tile * 16 + ((lane >> 4) << 3);
#pragma unroll
  for (int v = 0; v < 8; ++v) {
    int rm = mbase + v;
    if (rm < N) {
      float* dst = h2 + rm * F2 + n0;
      dst[0]  = c0[v];
      dst[16] = c1[v];
      dst[32] = c2[v];
      dst[48] = c3[v];
    }
  }
}

// ---------------- layer-2 edge aggregation: out[c,:] += norm * h2[r,:] ----------------
// 64 consecutive threads per edge -> coalesced 256B gather + coalesced atomics.
__global__ void gnn_scatter_h2(const int* __restrict__ row, const int* __restrict__ col,
                               const float* __restrict__ dinv, const float* __restrict__ h2,
                               float* __restrict__ out, int E) {
  int i = blockIdx.x * blockDim.x + threadIdx.x;
  int stride = gridDim.x * blockDim.x;
  int total = E * F2;
  for (int j = i; j < total; j += stride) {
    int e = j >> 6;
    int f = j & 63;
    int r = row[e], c = col[e];
    unsafeAtomicAdd(&out[c * F2 + f], dinv[r] * dinv[c] * h2[r * F2 + f]);
  }
}

// ---------------- self-loops + bias + relu ----------------
__global__ void gnn_finalize_out(const float* __restrict__ dinv, const float* __restrict__ h2,
                                 const float* __restrict__ b2, float* __restrict__ out, int N) {
  int i = blockIdx.x * blockDim.x + threadIdx.x;
  int total = N * F2;
  if (i < total) {
    int n = i >> 6;
    int f = i & 63;
    float v = out[i] + dinv[n] * dinv[n] * h2[i] + b2[f];
    out[i] = v > 0.0f ? v : 0.0f;
  }
}

extern "C" void kernel_launch(void* const* d_in, const int* in_sizes, int n_in,
                              void* d_out, int out_size, void* d_ws, size_t ws_size,
                              hipStream_t stream) {
  const int*   edge = (const int*)d_in[0];
  const int    E    = in_sizes[0] / 2;
  const int*   row  = edge;
  const int*   col  = edge + E;
  const float* W1   = (const float*)d_in[1];
  const float* b1   = (const float*)d_in[2];
  const float* W2   = (const float*)d_in[3];
  const float* b2   = (const float*)d_in[4];
  const int    N    = out_size / F2;

  float* ws   = (float*)d_ws;
  float* dinv = ws;            // N floats (in-degree -> dinv, in place)
  float* xdeg = ws + N;        // N floats (out-degree)
  float* s    = ws + 2 * N;    // N floats (aggregated layer-1 scalar)
  float* h2   = ws + 3 * N;    // N*64 floats (pre-aggregation layer-2 features)
  float* out  = (float*)d_out; // N*64 floats

  const int T = 256;

  // zero accumulators (h2 is fully overwritten, no need)
  gnn_zero<<<(3 * N + T - 1) / T, T, 0, stream>>>(ws, 3 * N);
  gnn_zero<<<(N * F2 + T - 1) / T, T, 0, stream>>>(out, N * F2);

  // degrees and normalization
  gnn_degrees<<<(E + T - 1) / T, T, 0, stream>>>(row, col, dinv, xdeg, E);
  gnn_dinv<<<(N + T - 1) / T, T, 0, stream>>>(dinv, N);

  // layer-1 (rank-1 W1) collapsed to scalar aggregation
  gnn_scatter_s<<<(E + T - 1) / T, T, 0, stream>>>(row, col, dinv, xdeg, s, E);
  gnn_finalize_s<<<(N + T - 1) / T, T, 0, stream>>>(dinv, xdeg, s, N);

  // fused relu(s (x) W1 + b1) @ W2 via f32 WMMA (128 rows per block)
  gnn_gemm_wmma<<<(N + 127) / 128, T, 0, stream>>>(s, W1, b1, W2, h2, N);

  // layer-2 aggregation + epilogue
  gnn_scatter_h2<<<((long long)E * F2 + T - 1) / T, T, 0, stream>>>(row, col, dinv, h2, out, E);
  gnn_finalize_out<<<(N * F2 + T - 1) / T, T, 0, stream>>>(dinv, h2, b2, out, N);
}